// NTXentLoss_87230785781949
// MI455X (gfx1250) — compile-verified
//
#include <hip/hip_runtime.h>
#include <hip/hip_bf16.h>
#include <math.h>

#define D 128           // feature dim (reference: 128)
#define TILE_PAD 132    // LDS row stride in floats (128 + 4 to dodge bank conflicts)

typedef __attribute__((ext_vector_type(2))) float v2f;
typedef __attribute__((ext_vector_type(8))) float v8f;

// virtual concatenated reps: row i in [0, 2N): i < N -> zjs[i], else zis[i-N]
__device__ __forceinline__ const float* reps_row(const float* zis, const float* zjs,
                                                 int i, int N) {
  return (i < N) ? (zjs + (size_t)i * D) : (zis + (size_t)(i - N) * D);
}

// ---------------------------------------------------------------------------
// Kernel 1: inverse row norms. One wave32 per row (128 floats -> float4/lane).
// ---------------------------------------------------------------------------
__global__ __launch_bounds__(256) void ntxent_norms(const float* __restrict__ zis,
                                                    const float* __restrict__ zjs,
                                                    float* __restrict__ inv, int N) {
  int wid  = (blockIdx.x * blockDim.x + threadIdx.x) >> 5;  // global wave id = row
  int lane = threadIdx.x & 31;
  if (wid >= 2 * N) return;
  const float4* row = (const float4*)reps_row(zis, zjs, wid, N);
  float4 v = row[lane];                      // 32 lanes * 4 = 128 elements
  float s = v.x * v.x + v.y * v.y + v.z * v.z + v.w * v.w;
  #pragma unroll
  for (int off = 16; off > 0; off >>= 1) s += __shfl_xor(s, off, 32);
  if (lane == 0) inv[wid] = 1.0f / fmaxf(sqrtf(s), 1e-8f);
}

// ---------------------------------------------------------------------------
// Kernel 2: fused sim GEMM (V_WMMA_F32_16X16X4_F32) + row-wise sum of exps.
// Block = 128 threads = 4 waves; each wave owns 16 rows; block owns 64 rows.
// Sweeps all 2N columns in 16-wide tiles staged through LDS.
//
// fp32 16x16x4 WMMA operand layouts (ISA 7.12.2):
//   A (16x4, 2 VGPR): lane m (0-15) holds K=0,1 ; lane m+16 holds K=2,3
//   B (4x16, 2 VGPR): lane n (0-15): VGPR0=K0,VGPR1=K1 ; lane n+16: K2,K3
//   C/D (16x16, 8 VGPR): VGPR v = row v (lanes 0-15) / row v+8 (lanes 16-31)
// ---------------------------------------------------------------------------
__global__ __launch_bounds__(128) void ntxent_denom(const float* __restrict__ zis,
                                                    const float* __restrict__ zjs,
                                                    const float* __restrict__ inv,
                                                    float* __restrict__ denom, int N) {
  __shared__ float ldsB[16 * TILE_PAD];     // one 16-col x 128-K panel

  const int twoN = 2 * N;
  const int tid  = threadIdx.x;
  const int lane = tid & 31;
  const int wave = tid >> 5;                // 0..3
  const int hi   = (lane >> 4) & 1;         // half-of-wave select
  const int m    = lane & 15;               // A row / B col within tile
  const int koff = hi ? 2 : 0;              // K sub-offset for this half

  const int row_base = blockIdx.x * 64 + wave * 16;
  const float* arow  = reps_row(zis, zjs, row_base + m, N);

  // Preload A panel: 32 K-tiles of (K=koff..koff+1) -> 64 VGPRs, reused 512x.
  v2f a[32];
  #pragma unroll
  for (int kt = 0; kt < 32; ++kt) {
    const float* p = arow + kt * 4 + koff;
    v2f t; t.x = p[0]; t.y = p[1];
    a[kt] = t;
  }

  // Per-lane inverse norms for the 8 rows this lane's C/D slots map to.
  float invr[8];
  #pragma unroll
  for (int v = 0; v < 8; ++v) invr[v] = inv[row_base + v + hi * 8];

  float acc[8];
  #pragma unroll
  for (int v = 0; v < 8; ++v) acc[v] = 0.0f;

  for (int ct = 0; ct < twoN / 16; ++ct) {
    const int col_base = ct * 16;

    __syncthreads();  // previous tile fully consumed
    // Cooperative stage of B panel: 16 cols x 128 K = 512 float4 loads.
    for (int idx = tid; idx < 16 * 32; idx += 128) {
      int n  = idx >> 5;          // column within tile
      int k4 = idx & 31;          // float4 index along K
      const float4* cp = (const float4*)reps_row(zis, zjs, col_base + n, N);
      float4 vv = cp[k4];
      float* dst = &ldsB[n * TILE_PAD + k4 * 4];
      dst[0] = vv.x; dst[1] = vv.y; dst[2] = vv.z; dst[3] = vv.w;
    }
    __syncthreads();

    // 16x16 tile of raw dot products via 32 chained fp32 WMMAs over K=128.
    v8f c = {};
    const float* bcol = &ldsB[m * TILE_PAD];   // this lane's B column (n = m)
    #pragma unroll
    for (int kt = 0; kt < 32; ++kt) {
      v2f b;
      b.x = bcol[kt * 4 + koff];
      b.y = bcol[kt * 4 + koff + 1];
      c = __builtin_amdgcn_wmma_f32_16x16x4_f32(
          /*neg_a=*/false, a[kt], /*neg_b=*/false, b,
          /*c_mod=*/(short)0, c, /*reuse_a=*/false, /*reuse_b=*/false);
    }

    // Fuse: cosine-scale, /T (=x2), exp, mask diagonal, accumulate per row.
    const int   gcol = col_base + m;
    const float invc = inv[gcol];
    #pragma unroll
    for (int v = 0; v < 8; ++v) {
      int   grow = row_base + v + hi * 8;
      float sim  = c[v] * invr[v] * invc;
      float e    = __expf(2.0f * sim);
      if (gcol == grow) e = 0.0f;     // exclude j == i only
      acc[v] += e;
    }
  }

  // Reduce each row's partial sums across its 16-lane half (fixed order).
  #pragma unroll
  for (int v = 0; v < 8; ++v) {
    float s = acc[v];
    s += __shfl_xor(s, 1, 32);
    s += __shfl_xor(s, 2, 32);
    s += __shfl_xor(s, 4, 32);
    s += __shfl_xor(s, 8, 32);
    if (m == 0) denom[row_base + v + hi * 8] = s;   // lanes 0 and 16 write
  }
}

// ---------------------------------------------------------------------------
// Kernel 3: positives + final deterministic reduction to the scalar loss.
// ---------------------------------------------------------------------------
__global__ __launch_bounds__(256) void ntxent_finalize(const float* __restrict__ zis,
                                                       const float* __restrict__ zjs,
                                                       const float* __restrict__ inv,
                                                       const float* __restrict__ denom,
                                                       float* __restrict__ out, int N) {
  __shared__ float red[256];
  const int tid = threadIdx.x;
  float sum = 0.0f;
  for (int i = tid; i < 2 * N; i += 256) {
    int partner = (i < N) ? (i + N) : (i - N);
    const float4* ri = (const float4*)reps_row(zis, zjs, i, N);
    const float4* rp = (const float4*)reps_row(zis, zjs, partner, N);
    float dot = 0.0f;
    #pragma unroll
    for (int k = 0; k < D / 4; ++k) {
      float4 x = ri[k], y = rp[k];
      dot += x.x * y.x + x.y * y.y + x.z * y.z + x.w * y.w;
    }
    float pos = dot * inv[i] * inv[partner];
    sum += logf(denom[i]) - 2.0f * pos;     // logsumexp_i - pos_i/T
  }
  red[tid] = sum;
  __syncthreads();
  #pragma unroll
  for (int s = 128; s > 0; s >>= 1) {
    if (tid < s) red[tid] += red[tid + s];
    __syncthreads();
  }
  if (tid == 0) out[0] = red[0] / (float)(2 * N);
}

// ---------------------------------------------------------------------------
extern "C" void kernel_launch(void* const* d_in, const int* in_sizes, int n_in,
                              void* d_out, int out_size, void* d_ws, size_t ws_size,
                              hipStream_t stream) {
  const float* zis = (const float*)d_in[0];
  const float* zjs = (const float*)d_in[1];
  const int N = in_sizes[0] / D;            // 4096
  const int twoN = 2 * N;

  float* inv   = (float*)d_ws;              // 2N floats
  float* denom = inv + twoN;                // 2N floats  (total 64 KB of ws)
  float* out   = (float*)d_out;

  // 1) inverse norms: 8 rows per 256-thread block
  ntxent_norms<<<twoN / 8, 256, 0, stream>>>(zis, zjs, inv, N);

  // 2) fused WMMA GEMM + exp-sum: 64 rows per 128-thread block
  ntxent_denom<<<twoN / 64, 128, 0, stream>>>(zis, zjs, inv, denom, N);

  // 3) scalar loss
  ntxent_finalize<<<1, 256, 0, stream>>>(zis, zjs, inv, denom, out, N);
}